// FastSpeech2Loss_80470507258140
// MI455X (gfx1250) — compile-verified
//
#include <hip/hip_runtime.h>
#include <hip/hip_bf16.h>
#include <cstdint>

#define B_      64
#define T_MEL   2000
#define T_SRC   256
#define N_MELS  80
#define D_ORIG  768

// ---------------------------------------------------------------------------
// Threefry2x32 (canonical, matches JAX's threefry PRNG)
// ---------------------------------------------------------------------------
__device__ __forceinline__ uint32_t rotl32(uint32_t x, uint32_t n) {
  return (x << n) | (x >> (32u - n));
}

__device__ __forceinline__ void threefry2x32(uint32_t k0, uint32_t k1,
                                             uint32_t x0, uint32_t x1,
                                             uint32_t& o0, uint32_t& o1) {
  uint32_t ks2 = k0 ^ k1 ^ 0x1BD11BDAu;
  x0 += k0; x1 += k1;
#define TFR(r) { x0 += x1; x1 = rotl32(x1, (r)); x1 ^= x0; }
  TFR(13) TFR(15) TFR(26) TFR(6)
  x0 += k1; x1 += ks2 + 1u;
  TFR(17) TFR(29) TFR(16) TFR(24)
  x0 += ks2; x1 += k0 + 2u;
  TFR(13) TFR(15) TFR(26) TFR(6)
  x0 += k0; x1 += k1 + 3u;
  TFR(17) TFR(29) TFR(16) TFR(24)
  x0 += k1; x1 += ks2 + 4u;
  TFR(13) TFR(15) TFR(26) TFR(6)
  x0 += ks2; x1 += k0 + 5u;
#undef TFR
  o0 = x0; o1 = x1;
}

// gumbel sample for flat element e of a 4096-element draw under key (ka,kb),
// exactly following jax.random.gumbel -> uniform -> random_bits lowering.
__device__ __forceinline__ float gumbel4096(uint32_t ka, uint32_t kb, uint32_t e) {
  uint32_t o0, o1, bits;
  if (e < 2048u) { threefry2x32(ka, kb, e, e + 2048u, o0, o1); bits = o0; }
  else           { threefry2x32(ka, kb, e - 2048u, e, o0, o1); bits = o1; }
  float f = __uint_as_float((bits >> 9) | 0x3f800000u) - 1.0f;   // [0,1)
  const float tiny = 1.1754944e-38f;
  float u = fmaxf(f * (1.0f - tiny) + tiny, tiny);
  return -logf(-logf(u));
}

// ---------------------------------------------------------------------------
// Reductions
// ---------------------------------------------------------------------------
__device__ __forceinline__ float waveReduce(float v) {
  #pragma unroll
  for (int off = 16; off > 0; off >>= 1) v += __shfl_down(v, off, 32);
  return v;
}

// blockDim.x == 256 (8 waves). Result valid in thread 0.
__device__ __forceinline__ float blockReduce(float v, float* sm8) {
  int lane = threadIdx.x & 31, wid = threadIdx.x >> 5;
  v = waveReduce(v);
  if (lane == 0) sm8[wid] = v;
  __syncthreads();
  v = (threadIdx.x < 8) ? sm8[threadIdx.x] : 0.0f;
  if (wid == 0) v = waveReduce(v);
  __syncthreads();
  return v;
}

// ---------------------------------------------------------------------------
// Kernel A: zero workspace accumulators
// ---------------------------------------------------------------------------
__global__ void zero_ws_kernel(float* w) {
  if (threadIdx.x < 16) w[threadIdx.x] = 0.0f;
}

// ---------------------------------------------------------------------------
// Kernel B: masked L1 reductions over the big mel tensors (bandwidth bound).
// acc[0] += sum |pred-tgt|*m ; acc[1] += sum |post-tgt|*m ; acc[2] += sum m
// ---------------------------------------------------------------------------
__global__ __launch_bounds__(256) void mel_kernel(const float* __restrict__ tgt,
                                                  const float* __restrict__ pred,
                                                  const float* __restrict__ post,
                                                  const unsigned char* __restrict__ mel_mask,
                                                  float* acc) {
  __shared__ float sm8[8];
  const int total4 = B_ * T_MEL * (N_MELS / 4);       // 2,560,000 float4s
  const int stride = gridDim.x * blockDim.x;
  const float4* t4 = (const float4*)tgt;
  const float4* p4 = (const float4*)pred;
  const float4* q4 = (const float4*)post;

  float s_mel = 0.0f, s_post = 0.0f, s_den = 0.0f;
  for (int g = blockIdx.x * blockDim.x + threadIdx.x; g < total4; g += stride) {
    int gn = g + stride;
    if (gn < total4) {                                 // CDNA5 global_prefetch_b8
      __builtin_prefetch((const void*)(t4 + gn), 0, 0);
      __builtin_prefetch((const void*)(p4 + gn), 0, 0);
      __builtin_prefetch((const void*)(q4 + gn), 0, 0);
    }
    int row = g / (N_MELS / 4);                        // (b, t_mel) row
    float m = mel_mask[row] ? 0.0f : 1.0f;             // mel_m = ~mel_masks
    float4 t = t4[g], p = p4[g], q = q4[g];
    s_mel  += m * (fabsf(p.x - t.x) + fabsf(p.y - t.y) +
                   fabsf(p.z - t.z) + fabsf(p.w - t.w));
    s_post += m * (fabsf(q.x - t.x) + fabsf(q.y - t.y) +
                   fabsf(q.z - t.z) + fabsf(q.w - t.w));
    s_den  += 4.0f * m;
  }
  float r0 = blockReduce(s_mel, sm8);
  float r1 = blockReduce(s_post, sm8);
  float r2 = blockReduce(s_den, sm8);
  if (threadIdx.x == 0) {
    atomicAdd(acc + 0, r0);
    atomicAdd(acc + 1, r1);
    atomicAdd(acc + 2, r2);
  }
}

// ---------------------------------------------------------------------------
// Kernel C: pitch / energy / duration masked MSE sums + src mask count
// ---------------------------------------------------------------------------
__global__ __launch_bounds__(256) void src_kernel(const float* __restrict__ pt,
                                                  const float* __restrict__ et,
                                                  const float* __restrict__ pp,
                                                  const float* __restrict__ ep,
                                                  const float* __restrict__ ldp,
                                                  const int* __restrict__ dt,
                                                  const unsigned char* __restrict__ smask,
                                                  float* acc) {
  __shared__ float sm8[8];
  const int n = B_ * T_SRC;
  float sp = 0.0f, se = 0.0f, sd = 0.0f, sden = 0.0f;
  for (int i = blockIdx.x * blockDim.x + threadIdx.x; i < n;
       i += gridDim.x * blockDim.x) {
    float m = smask[i] ? 0.0f : 1.0f;                  // src_m = ~src_masks
    float dpi = pp[i] - pt[i];
    float dei = ep[i] - et[i];
    float ldt = logf((float)dt[i] + 1.0f);
    float ddi = ldp[i] - ldt;
    sp += m * dpi * dpi;
    se += m * dei * dei;
    sd += m * ddi * ddi;
    sden += m;
  }
  float r0 = blockReduce(sp, sm8);
  float r1 = blockReduce(se, sm8);
  float r2 = blockReduce(sd, sm8);
  float r3 = blockReduce(sden, sm8);
  if (threadIdx.x == 0) {
    atomicAdd(acc + 3, r0);
    atomicAdd(acc + 4, r1);
    atomicAdd(acc + 5, r2);
    atomicAdd(acc + 6, r3);
  }
}

// ---------------------------------------------------------------------------
// Kernel D: style L1 sum
// ---------------------------------------------------------------------------
__global__ __launch_bounds__(256) void style_kernel(const float* __restrict__ ref,
                                                    const float* __restrict__ pred,
                                                    float* acc) {
  __shared__ float sm8[8];
  const int n = B_ * 256;
  float s = 0.0f;
  for (int i = blockIdx.x * blockDim.x + threadIdx.x; i < n;
       i += gridDim.x * blockDim.x)
    s += fabsf(pred[i] - ref[i]);
  float r = blockReduce(s, sm8);
  if (threadIdx.x == 0) atomicAdd(acc + 7, r);
}

// ---------------------------------------------------------------------------
// Kernel E: triplet loss per 128-dim chunk (one block per chunk, 2 waves).
// Gram matrix G = E * E^T via V_WMMA_F32_16X16X4_F32; distances recovered as
// ||a-p+eps||^2 = Gii - 2Gip + Gpp + 2eps(Si - Sp) + D*eps^2.
// ---------------------------------------------------------------------------
#if defined(__gfx1250__) && __has_builtin(__builtin_amdgcn_wmma_f32_16x16x4_f32)
#define HAVE_WMMA_F32X4 1
typedef float v2f __attribute__((ext_vector_type(2)));
typedef float v8f __attribute__((ext_vector_type(8)));
#else
#define HAVE_WMMA_F32X4 0
#endif

__global__ __launch_bounds__(64) void triplet_kernel(const float* __restrict__ orig,
                                                     const int* __restrict__ emotions,
                                                     float* acc) {
  __shared__ float E[64][128];
  __shared__ float G[64][64];
  __shared__ float S[64];
  __shared__ int   emo[64];
  __shared__ float perArr[64];
  __shared__ int   validArr[64];

  const int idx = blockIdx.x;          // chunk 0..5
  const int tid = threadIdx.x;         // 0..63

  // Load chunk rows + row sums.
  {
    const float* src = orig + tid * D_ORIG + idx * 128;
    float s = 0.0f;
    #pragma unroll 4
    for (int k = 0; k < 128; ++k) { float v = src[k]; E[tid][k] = v; s += v; }
    S[tid] = s;
    emo[tid] = emotions[tid];
  }
  __syncthreads();

#if HAVE_WMMA_F32X4
  {
    const int lane = tid & 31;
    const int wave = tid >> 5;
    const int half = lane >> 4;        // 0: K pair {0,1}, 1: K pair {2,3}
    const int l16  = lane & 15;
    for (int t = wave * 8; t < wave * 8 + 8; ++t) {
      const int mt = t >> 2, nt = t & 3;
      v8f c = {0.f, 0.f, 0.f, 0.f, 0.f, 0.f, 0.f, 0.f};
      const int rA = mt * 16 + l16;
      const int rB = nt * 16 + l16;
      for (int kb = 0; kb < 128; kb += 4) {
        const int k0 = kb + half * 2;
        v2f a; a.x = E[rA][k0]; a.y = E[rA][k0 + 1];
        v2f b; b.x = E[rB][k0]; b.y = E[rB][k0 + 1];
        c = __builtin_amdgcn_wmma_f32_16x16x4_f32(
                false, a, false, b, (short)0, c, false, false);
      }
      // D layout: VGPR r -> lanes0-15: (M=mt*16+r, N=nt*16+lane); lanes16-31: M+8
      const int gm = mt * 16 + half * 8;
      const int gn = nt * 16 + l16;
      #pragma unroll
      for (int r = 0; r < 8; ++r) G[gm + r][gn] = c[r];
    }
  }
#else
  for (int j = 0; j < 64; ++j) {
    float d = 0.0f;
    #pragma unroll 4
    for (int k = 0; k < 128; ++k) d += E[tid][k] * E[j][k];
    G[tid][j] = d;
  }
#endif
  __syncthreads();

  // JAX key derivation: tkey = key(42) = (0,42); k = fold_in(tkey, idx);
  // k1,k2 = split(k).
  uint32_t fk0, fk1;
  threefry2x32(0u, 42u, 0u, (uint32_t)idx, fk0, fk1);
  uint32_t a0, b0, a1, b1;
  threefry2x32(fk0, fk1, 0u, 2u, a0, b0);
  threefry2x32(fk0, fk1, 1u, 3u, a1, b1);
  const uint32_t k1a = a0, k1b = a1;   // positive-sampling key
  const uint32_t k2a = b0, k2b = b1;   // negative-sampling key

  // Row tid: gumbel-argmax positive / negative selection.
  const int li = emo[tid];
  float bestP = -3.0e38f, bestN = -3.0e38f;
  int   pi = 0, ni = 0;
  bool  anyS = false, anyD = false;
  for (int j = 0; j < 64; ++j) {
    const bool sm = (emo[j] == li) && (j != tid);
    const bool df = (emo[j] != li);
    const uint32_t e = (uint32_t)(tid * 64 + j);
    const float g1 = sm ? gumbel4096(k1a, k1b, e) : -1.0e9f;
    const float g2 = df ? gumbel4096(k2a, k2b, e) : -1.0e9f;
    if (g1 > bestP) { bestP = g1; pi = j; }
    if (g2 > bestN) { bestN = g2; ni = j; }
    anyS |= sm; anyD |= df;
  }

  const float eps = 1.0e-6f;
  const float cst = 128.0f * eps * eps;
  float d2p = G[tid][tid] + G[pi][pi] - 2.0f * G[tid][pi]
              + 2.0f * eps * (S[tid] - S[pi]) + cst;
  float d2n = G[tid][tid] + G[ni][ni] - 2.0f * G[tid][ni]
              + 2.0f * eps * (S[tid] - S[ni]) + cst;
  const float dp = sqrtf(fmaxf(d2p, 0.0f));
  const float dn = sqrtf(fmaxf(d2n, 0.0f));
  const float per = fmaxf(dp - dn + 1.0f, 0.0f);
  const bool valid = anyS && anyD;
  perArr[tid]   = valid ? per : 0.0f;
  validArr[tid] = valid ? 1 : 0;
  __syncthreads();

  if (tid == 0) {
    float s = 0.0f; int c = 0;
    for (int j = 0; j < 64; ++j) { s += perArr[j]; c += validArr[j]; }
    const float loss = (c > 0) ? (s / (float)c) : 0.0f;
    const float beta = ((idx & 1) == 0) ? 0.5f : 0.1f;
    atomicAdd(acc + 8, beta * loss);
  }
}

// ---------------------------------------------------------------------------
// Kernel F: combine scalars
// ---------------------------------------------------------------------------
__global__ void combine_kernel(const float* __restrict__ w,
                               const float* __restrict__ guided,
                               const float* __restrict__ vq,
                               float* __restrict__ out) {
  if (threadIdx.x == 0 && blockIdx.x == 0) {
    const float mel    = w[0] / fmaxf(w[2], 1.0f);
    const float post   = w[1] / fmaxf(w[2], 1.0f);
    const float pitch  = w[3] / fmaxf(w[6], 1.0f);
    const float energy = w[4] / fmaxf(w[6], 1.0f);
    const float dur    = w[5] / fmaxf(w[6], 1.0f);
    const float style  = (w[7] / (float)(B_ * 256)) * 10.0f;
    const float g = guided[0];
    const float v = vq[0];
    const float cls = w[8];
    const float total = mel + post + dur + pitch + energy + (style + g) + v + cls;
    out[0] = total; out[1] = mel;  out[2] = post;  out[3] = pitch;
    out[4] = energy; out[5] = dur; out[6] = style; out[7] = g;
    out[8] = v;      out[9] = cls;
  }
}

// ---------------------------------------------------------------------------
// Launcher
// ---------------------------------------------------------------------------
extern "C" void kernel_launch(void* const* d_in, const int* in_sizes, int n_in,
                              void* d_out, int out_size, void* d_ws, size_t ws_size,
                              hipStream_t stream) {
  const float* mel_t   = (const float*)d_in[0];
  const float* pitch_t = (const float*)d_in[1];
  const float* energy_t= (const float*)d_in[2];
  const float* mel_p   = (const float*)d_in[3];
  const float* post_p  = (const float*)d_in[4];
  const float* pitch_p = (const float*)d_in[5];
  const float* energy_p= (const float*)d_in[6];
  const float* logdur_p= (const float*)d_in[7];
  const float* sref    = (const float*)d_in[8];
  const float* spred   = (const float*)d_in[9];
  const float* guided  = (const float*)d_in[10];
  const float* vq      = (const float*)d_in[11];
  const float* orig    = (const float*)d_in[12];
  const int*   dur_t   = (const int*)d_in[13];
  const int*   emo     = (const int*)d_in[14];
  const unsigned char* src_mask = (const unsigned char*)d_in[15];
  const unsigned char* mel_mask = (const unsigned char*)d_in[16];
  float* out = (float*)d_out;
  float* w   = (float*)d_ws;

  zero_ws_kernel<<<1, 32, 0, stream>>>(w);
  mel_kernel<<<2048, 256, 0, stream>>>(mel_t, mel_p, post_p, mel_mask, w);
  src_kernel<<<32, 256, 0, stream>>>(pitch_t, energy_t, pitch_p, energy_p,
                                     logdur_p, dur_t, src_mask, w);
  style_kernel<<<32, 256, 0, stream>>>(sref, spred, w);
  triplet_kernel<<<6, 64, 0, stream>>>(orig, emo, w);
  combine_kernel<<<1, 32, 0, stream>>>(w, guided, vq, out);
}